// DynamicNeuralTuringMachine_7713761263952
// MI455X (gfx1250) — compile-verified
//
#include <hip/hip_runtime.h>
#include <math.h>

typedef __attribute__((ext_vector_type(2))) float v2f;
typedef __attribute__((ext_vector_type(8))) float v8f;

#define B_  32
#define T_  32
#define F_  64
#define H_  256
#define N_  1024
#define A_  64
#define C_  192
#define M_  256          // A_ + C_
#define TB_ (T_ * B_)    // 1024
#define NCHUNK_ 8
#define CHUNKN_ (N_ / NCHUNK_)   // 128

// ---------------------------------------------------------------------------
// Generic f32 WMMA GEMM: D[m,n] = sum_k A[m*lda+k] * Bm[k*ldb+n] (+ bias[m])
// One wave (32 threads) per 16x16 output tile. grid = (M/16, N/16).
// V_WMMA_F32_16X16X4_F32 lane layout: A: lane l -> row = l&15, VGPR0/1 hold
// K = 2*(l>>4) + {0,1}. B mirrored: col = l&15, same K split.
// C/D: VGPR v, lanes 0-15 -> M=v, lanes 16-31 -> M=v+8, N = l&15.
// ---------------------------------------------------------------------------
__global__ __launch_bounds__(32) void wmma_gemm(const float* __restrict__ A,
                                                const float* __restrict__ Bm,
                                                const float* __restrict__ bias,
                                                float* __restrict__ D,
                                                int K, int lda, int ldb, int ldd,
                                                int addBias) {
  const int lane = threadIdx.x & 31;
  const int tm = blockIdx.x * 16;
  const int tn = blockIdx.y * 16;
  const int row = lane & 15;            // A row / B col / D col
  const int kh  = (lane >> 4) << 1;     // K sub-offset 0 or 2
  v8f acc = {0.f, 0.f, 0.f, 0.f, 0.f, 0.f, 0.f, 0.f};
  const float* Ap = A + (size_t)(tm + row) * lda + kh;
  const float* Bp = Bm + (size_t)kh * ldb + tn + row;
  for (int k = 0; k < K; k += 4) {
    v2f a, b;
    a.x = Ap[k];
    a.y = Ap[k + 1];
    b.x = Bp[(size_t)k * ldb];
    b.y = Bp[(size_t)k * ldb + ldb];
    acc = __builtin_amdgcn_wmma_f32_16x16x4_f32(false, a, false, b,
                                                (short)0, acc, false, false);
  }
  const int rb = (lane >> 4) << 3;
#pragma unroll
  for (int v = 0; v < 8; ++v) {
    int m = tm + rb + v;
    float val = acc[v];
    if (addBias) val += bias[m];
    D[(size_t)m * ldd + tn + row] = val;
  }
}

// ---------------------------------------------------------------------------
__global__ void zero_kernel(float* p, int n) {
  int i = blockIdx.x * blockDim.x + threadIdx.x;
  if (i < n) p[i] = 0.f;
}

// xs[t][f][b] = batch[(f*B+b)/F][t][(f*B+b)%F]  (raw reshape per reference)
// stored as XS2[f][t*B+b]  -> [F_, TB_]
__global__ void reshape_kernel(const float* __restrict__ batch,
                               float* __restrict__ XS2) {
  int idx = blockIdx.x * blockDim.x + threadIdx.x;  // t*F*B + f*B + b
  if (idx >= T_ * F_ * B_) return;
  int t = idx / (F_ * B_);
  int r = idx - t * (F_ * B_);
  int f = r / B_;
  int b = r - f * B_;
  int l  = f * B_ + b;
  int bb = l / F_;
  int ff = l - bb * F_;
  XS2[(size_t)f * TB_ + t * B_ + b] = batch[((size_t)bb * T_ + t) * F_ + ff];
}

// beta[b] = softplus(u . h[:,b]) + 1
__global__ void beta_kernel(const float* __restrict__ u,
                            const float* __restrict__ h,
                            float* __restrict__ beta) {
  int b = threadIdx.x;
  if (b >= B_) return;
  float acc = 0.f;
  for (int k = 0; k < H_; ++k) acc += u[k] * h[k * B_ + b];
  float sp = (acc > 20.f) ? acc : log1pf(expf(acc));
  beta[b] = sp + 1.f;
}

// cand = relu(candH + CIall[:, t*B:(t+1)*B])
__global__ void cand_kernel(const float* __restrict__ candH,
                            const float* __restrict__ CIall,
                            float* __restrict__ cand, int t) {
  int idx = blockIdx.x * blockDim.x + threadIdx.x;  // c*B + b
  if (idx >= C_ * B_) return;
  int c = idx >> 5;
  int b = idx & 31;
  float v = candH[idx] + CIall[(size_t)c * TB_ + t * B_ + b];
  cand[idx] = fmaxf(v, 0.f);
}

// sim[n,b] = beta[b] * ( addr[n,:64].q[0:64,b] + content[b,n,:].q[64:256,b] )
__global__ void sim_kernel(const float* __restrict__ addr,
                           const float* __restrict__ content,
                           const float* __restrict__ q,
                           const float* __restrict__ beta,
                           float* __restrict__ sim) {
  int tid = blockIdx.x * blockDim.x + threadIdx.x;  // n*B + b
  if (tid >= N_ * B_) return;
  int n = tid >> 5;
  int b = tid & 31;
  float acc = 0.f;
  const float4* an4 = (const float4*)(addr + (size_t)n * A_);
#pragma unroll 4
  for (int m4 = 0; m4 < A_ / 4; ++m4) {
    float4 v = an4[m4];
    int m = m4 * 4;
    acc += v.x * q[m * B_ + b] + v.y * q[(m + 1) * B_ + b] +
           v.z * q[(m + 2) * B_ + b] + v.w * q[(m + 3) * B_ + b];
  }
  const float4* cn4 = (const float4*)(content + ((size_t)b * N_ + n) * C_);
#pragma unroll 4
  for (int c4 = 0; c4 < C_ / 4; ++c4) {
    float4 v = cn4[c4];
    int c = A_ + c4 * 4;
    acc += v.x * q[c * B_ + b] + v.y * q[(c + 1) * B_ + b] +
           v.z * q[(c + 2) * B_ + b] + v.w * q[(c + 3) * B_ + b];
  }
  sim[tid] = beta[b] * acc;
}

// column softmax over N (axis 0): one block per batch column b
__global__ void softmax_kernel(const float* __restrict__ sim,
                               float* __restrict__ w) {
  __shared__ float red[256];
  int b = blockIdx.x;
  int tid = threadIdx.x;
  float vals[4];
  float local = -3.4e38f;
#pragma unroll
  for (int i = 0; i < 4; ++i) {
    vals[i] = sim[(tid + i * 256) * B_ + b];
    local = fmaxf(local, vals[i]);
  }
  red[tid] = local;
  __syncthreads();
  for (int s = 128; s > 0; s >>= 1) {
    if (tid < s) red[tid] = fmaxf(red[tid], red[tid + s]);
    __syncthreads();
  }
  float mx = red[0];
  __syncthreads();
  float ls = 0.f;
#pragma unroll
  for (int i = 0; i < 4; ++i) {
    vals[i] = expf(vals[i] - mx);
    ls += vals[i];
  }
  red[tid] = ls;
  __syncthreads();
  for (int s = 128; s > 0; s >>= 1) {
    if (tid < s) red[tid] += red[tid + s];
    __syncthreads();
  }
  float inv = 1.f / red[0];
#pragma unroll
  for (int i = 0; i < 4; ++i) w[(tid + i * 256) * B_ + b] = vals[i] * inv;
}

// Partial reading reduction over an n-chunk:
// partial[chunk][m][b] = sum_{n in chunk} full[b,n,m]*w[n,b]
// grid = (B_, NCHUNK_), block = 256 (m)
__global__ void read_partial_kernel(const float* __restrict__ addr,
                                    const float* __restrict__ content,
                                    const float* __restrict__ w,
                                    float* __restrict__ partial) {
  int b = blockIdx.x;
  int chunk = blockIdx.y;
  int m = threadIdx.x;  // 0..255
  int n0 = chunk * CHUNKN_;
  float acc = 0.f;
  if (m < A_) {
    const float* ap = addr + (size_t)n0 * A_ + m;
    for (int i = 0; i < CHUNKN_; ++i)
      acc += ap[(size_t)i * A_] * w[(n0 + i) * B_ + b];
  } else {
    const float* cb = content + ((size_t)b * N_ + n0) * C_ + (m - A_);
    for (int i = 0; i < CHUNKN_; ++i)
      acc += cb[(size_t)i * C_] * w[(n0 + i) * B_ + b];
  }
  partial[((size_t)chunk * M_ + m) * B_ + b] = acc;
}

// reading[m,b] = sum_chunk partial[chunk][m][b]
__global__ void read_finalize_kernel(const float* __restrict__ partial,
                                     float* __restrict__ reading) {
  int idx = blockIdx.x * blockDim.x + threadIdx.x;  // m*B + b
  if (idx >= M_ * B_) return;
  float acc = 0.f;
#pragma unroll
  for (int c = 0; c < NCHUNK_; ++c) acc += partial[(size_t)c * M_ * B_ + idx];
  reading[idx] = acc;
}

// content[b,n,c] += w[n,b]*cand[c,b].  grid = B_*N_ blocks of C_ threads.
__global__ void content_update_kernel(float* __restrict__ content,
                                      const float* __restrict__ w,
                                      const float* __restrict__ cand) {
  int blk = blockIdx.x;          // b*N + n
  int b = blk >> 10;             // N_ = 1024
  int n = blk & (N_ - 1);
  int c = threadIdx.x;           // 0..191
  float wv = w[n * B_ + b];
  size_t off = (size_t)blk * C_ + c;
  content[off] += wv * cand[c * B_ + b];
}

// GRU gates + h update; writes h (in place) and hs[t]
__global__ void gates_kernel(const float* __restrict__ GIall,
                             const float* __restrict__ bi,
                             const float* __restrict__ GH,
                             const float* __restrict__ GM,
                             float* __restrict__ h,
                             float* __restrict__ hs, int t) {
  int idx = blockIdx.x * blockDim.x + threadIdx.x;  // hh*B + b
  if (idx >= H_ * B_) return;
  int hh = idx >> 5;
  int b = idx & 31;
  int col = t * B_ + b;
  float gi0 = GIall[(size_t)(0 * H_ + hh) * TB_ + col] + bi[0 * H_ + hh];
  float gi1 = GIall[(size_t)(1 * H_ + hh) * TB_ + col] + bi[1 * H_ + hh];
  float gi2 = GIall[(size_t)(2 * H_ + hh) * TB_ + col] + bi[2 * H_ + hh];
  float gh0 = GH[(0 * H_ + hh) * B_ + b];
  float gh1 = GH[(1 * H_ + hh) * B_ + b];
  float gh2 = GH[(2 * H_ + hh) * B_ + b];
  float gm0 = GM[(0 * H_ + hh) * B_ + b];
  float gm1 = GM[(1 * H_ + hh) * B_ + b];
  float gm2 = GM[(2 * H_ + hh) * B_ + b];
  float r = 1.f / (1.f + expf(-(gi0 + gh0 + gm0)));
  float z = 1.f / (1.f + expf(-(gi1 + gh1 + gm1)));
  float nn = tanhf(gi2 + gm2 + r * gh2);
  float hold = h[idx];
  float hn = (1.f - z) * nn + z * hold;
  h[idx] = hn;
  hs[(size_t)t * H_ * B_ + idx] = hn;
}

// outs[t] = log_softmax(Wout @ h + bout) over the 10 rows
__global__ void out_kernel(const float* __restrict__ Wout,
                           const float* __restrict__ bout,
                           const float* __restrict__ h,
                           float* __restrict__ outs, int t) {
  __shared__ float lg[10 * 32];
  __shared__ float lse[32];
  int tid = threadIdx.x;  // 0..319
  int o = tid / 32;
  int b = tid & 31;
  float acc = bout[o];
  for (int k = 0; k < H_; ++k) acc += Wout[o * H_ + k] * h[k * B_ + b];
  lg[o * 32 + b] = acc;
  __syncthreads();
  if (o == 0) {
    float mx = -3.4e38f;
    for (int i = 0; i < 10; ++i) mx = fmaxf(mx, lg[i * 32 + b]);
    float s = 0.f;
    for (int i = 0; i < 10; ++i) s += expf(lg[i * 32 + b] - mx);
    lse[b] = mx + logf(s);
  }
  __syncthreads();
  outs[(size_t)t * 10 * B_ + o * 32 + b] = lg[o * 32 + b] - lse[b];
}

// ---------------------------------------------------------------------------
extern "C" void kernel_launch(void* const* d_in, const int* in_sizes, int n_in,
                              void* d_out, int out_size, void* d_ws,
                              size_t ws_size, hipStream_t stream) {
  const float* batch = (const float*)d_in[0];
  const float* Wi    = (const float*)d_in[1];   // [768,64]
  const float* bi    = (const float*)d_in[2];   // [768]
  const float* Wh    = (const float*)d_in[3];   // [768,256]
  const float* bh    = (const float*)d_in[4];   // [768]
  const float* Wm    = (const float*)d_in[5];   // [768,256]
  const float* bm    = (const float*)d_in[6];   // [768]
  const float* Wout  = (const float*)d_in[7];   // [10,256]
  const float* bout  = (const float*)d_in[8];   // [10]
  const float* addr  = (const float*)d_in[9];   // [1024,64]
  const float* Wq    = (const float*)d_in[10];  // [256,256]
  const float* bq    = (const float*)d_in[11];  // [256]
  const float* u_sh  = (const float*)d_in[12];  // [256]
  const float* Wch   = (const float*)d_in[13];  // [192,256]
  const float* Wci   = (const float*)d_in[14];  // [192,64]

  float* ws = (float*)d_ws;
  float* XS2     = ws;                    // 65536
  float* GIall   = XS2 + 65536;           // 786432
  float* CIall   = GIall + 786432;        // 196608
  float* content = CIall + 196608;        // 6291456
  float* h       = content + 6291456;     // 8192 (contiguous after content)
  float* q       = h + 8192;              // 8192
  float* GH      = q + 8192;              // 24576
  float* GM      = GH + 24576;            // 24576
  float* candH   = GM + 24576;            // 6144
  float* cand    = candH + 6144;          // 6144
  float* beta    = cand + 6144;           // 32 (pad to 256)
  float* sim     = beta + 256;            // 32768
  float* w       = sim + 32768;           // 32768
  float* reading = w + 32768;             // 8192
  float* partial = reading + 8192;        // NCHUNK_*M_*B_ = 65536

  float* hs_out   = (float*)d_out;             // [T,H,B]
  float* outs_out = hs_out + T_ * H_ * B_;     // [T,10,B]

  // zero state (content + h are contiguous)
  {
    int n = N_ * C_ * B_ + H_ * B_;
    zero_kernel<<<(n + 255) / 256, 256, 0, stream>>>(content, n);
  }
  reshape_kernel<<<(T_ * F_ * B_ + 255) / 256, 256, 0, stream>>>(batch, XS2);

  // hoisted input GEMMs: GIall = Wi @ XS2, CIall = Wci @ XS2
  wmma_gemm<<<dim3(48, 64), 32, 0, stream>>>(Wi, XS2, nullptr, GIall,
                                             F_, F_, TB_, TB_, 0);
  wmma_gemm<<<dim3(12, 64), 32, 0, stream>>>(Wci, XS2, nullptr, CIall,
                                             F_, F_, TB_, TB_, 0);

  for (int t = 0; t < T_; ++t) {
    // q = Wq@h + bq ; candH = Wch@h ; GH = Wh@h + bh
    wmma_gemm<<<dim3(16, 2), 32, 0, stream>>>(Wq, h, bq, q, H_, H_, B_, B_, 1);
    wmma_gemm<<<dim3(12, 2), 32, 0, stream>>>(Wch, h, nullptr, candH,
                                              H_, H_, B_, B_, 0);
    wmma_gemm<<<dim3(48, 2), 32, 0, stream>>>(Wh, h, bh, GH, H_, H_, B_, B_, 1);
    beta_kernel<<<1, 32, 0, stream>>>(u_sh, h, beta);
    cand_kernel<<<(C_ * B_ + 255) / 256, 256, 0, stream>>>(candH, CIall, cand, t);
    sim_kernel<<<(N_ * B_ + 255) / 256, 256, 0, stream>>>(addr, content, q,
                                                          beta, sim);
    softmax_kernel<<<B_, 256, 0, stream>>>(sim, w);
    // reading (reads content BEFORE update), split over n-chunks for occupancy
    read_partial_kernel<<<dim3(B_, NCHUNK_), 256, 0, stream>>>(addr, content,
                                                               w, partial);
    read_finalize_kernel<<<(M_ * B_ + 255) / 256, 256, 0, stream>>>(partial,
                                                                    reading);
    // content += w (outer) cand, fully parallel
    content_update_kernel<<<B_ * N_, C_, 0, stream>>>(content, w, cand);
    // GM = Wm @ reading + bm
    wmma_gemm<<<dim3(48, 2), 32, 0, stream>>>(Wm, reading, bm, GM,
                                              M_, M_, B_, B_, 1);
    gates_kernel<<<(H_ * B_ + 255) / 256, 256, 0, stream>>>(GIall, bi, GH, GM,
                                                            h, hs_out, t);
    out_kernel<<<1, 320, 0, stream>>>(Wout, bout, h, outs_out, t);
  }
}